// SetAbstraction_60610578481728
// MI455X (gfx1250) — compile-verified
//
#include <hip/hip_runtime.h>
#include <hip/hip_bf16.h>

typedef float v2f __attribute__((ext_vector_type(2)));
typedef float v8f __attribute__((ext_vector_type(8)));

#define NPTS   100000
#define MCENT  25000
#define EEDGE  800000
#define CIN    64
#define CMID   64
#define COUT   128
#define K1PAD  68            // 3 + 64 padded to multiple of 4
#define BN_EPS 1e-5f
#define PASS_BLOCKS 6250     // EEDGE / 128 (8 waves * 16 rows)

__device__ __forceinline__ v8f v8zero() {
  v8f v = {0.f,0.f,0.f,0.f,0.f,0.f,0.f,0.f};
  return v;
}

// D = A(16x4 f32) * B(4x16 f32) + C, exact fp32 on the matrix pipe.
__device__ __forceinline__ v8f wmma4(v2f a, v2f b, v8f c) {
  return __builtin_amdgcn_wmma_f32_16x16x4_f32(false, a, false, b, (short)0, c,
                                               false, false);
}

// Order-preserving float<->uint map so segment max/min use integer atomics
// (deterministic, unlike float atomic adds).
__device__ __forceinline__ unsigned ordf(float f) {
  unsigned u = __float_as_uint(f);
  return (u & 0x80000000u) ? ~u : (u | 0x80000000u);
}
__device__ __forceinline__ float unordf(unsigned u) {
  return __uint_as_float((u & 0x80000000u) ? (u ^ 0x80000000u) : ~u);
}

// Stage a KxC weight matrix into LDS as K-pair-interleaved float2 so a lane's
// B fragment (W[k][col], W[k+1][col]) is a single aligned 8B ds_load_b64.
//   Wp[kp*C + col] = { W[2kp][col], W[2kp+1][col] }   (rows >= krows are 0)
__device__ __forceinline__ void stage_pairs(v2f* __restrict__ Wp,
                                            const float* __restrict__ W,
                                            int kpairs, int C, int krows) {
  for (int t = threadIdx.x; t < kpairs * C; t += blockDim.x) {
    const int kp = t / C, col = t - kp * C;
    const int k0 = 2 * kp, k1 = k0 + 1;
    v2f w;
    w.x = (k0 < krows) ? W[k0 * C + col] : 0.f;
    w.y = (k1 < krows) ? W[k1 * C + col] : 0.f;
    Wp[t] = w;
  }
}

// ---------------------------------------------------------------- init
__global__ void __launch_bounds__(256) k_init(unsigned* pmax, unsigned* pmin, int n) {
  int i = blockIdx.x * blockDim.x + threadIdx.x;
  if (i < n) { pmax[i] = 0u; pmin[i] = 0xFFFFFFFFu; }   // below ord(-inf)/above ord(+inf)
}

// ------------------------------------------------- skip conv + new_p
// identity[m] = x[idx[m]] @ Wskip + bskip ; new_p[m] = p[idx[m]]
__global__ void __launch_bounds__(256) k_skip(
    const float* __restrict__ p, const float* __restrict__ x,
    const int* __restrict__ idx,
    const float* __restrict__ Wskip, const float* __restrict__ bskip,
    float* __restrict__ identity, float* __restrict__ newp) {
  extern __shared__ float smem[];
  v2f* Wp = (v2f*)smem;                                 // [CIN/2][COUT] pairs
  stage_pairs(Wp, Wskip, CIN / 2, COUT, CIN);
  __syncthreads();

  const int wid = threadIdx.x >> 5, lane = threadIdx.x & 31;
  const int half = lane >> 4, r = lane & 15;
  const int row0 = (blockIdx.x * 8 + wid) * 16;
  const int myrow = row0 + r;
  const bool valid = myrow < MCENT;
  const int srcrow = valid ? idx[myrow] : 0;
  if (half == 0 && valid) {
    newp[myrow * 3 + 0] = p[srcrow * 3 + 0];
    newp[myrow * 3 + 1] = p[srcrow * 3 + 1];
    newp[myrow * 3 + 2] = p[srcrow * 3 + 2];
  }
  const float* xr = x + (size_t)srcrow * CIN;
  __builtin_prefetch(xr, 0, 1);

  v8f acc[8];
  #pragma unroll
  for (int n = 0; n < 8; ++n) acc[n] = v8zero();

  #pragma unroll
  for (int kk = 0; kk < CIN / 4; ++kk) {
    const int kp = 2 * kk + half;                       // pair index (k = 2*kp)
    const v2f a = *(const v2f*)(xr + 2 * kp);           // 8B-aligned global b64
    #pragma unroll
    for (int n = 0; n < 8; ++n)
      acc[n] = wmma4(a, Wp[kp * COUT + n * 16 + r], acc[n]);
  }
  #pragma unroll
  for (int n = 0; n < 8; ++n) {
    const int col = n * 16 + r;
    const float bb = bskip[col];
    #pragma unroll
    for (int v = 0; v < 8; ++v) {
      const int row = row0 + v + 8 * half;
      if (row < MCENT) identity[(size_t)row * COUT + col] = acc[n][v] + bb;
    }
  }
}

// -------------------------------------------- pass 1: BN1 statistics
// h_pre = feat @ W1 + b1; accumulate per-channel sum / sum^2 partials.
__global__ void __launch_bounds__(256) k_pass1(
    const float* __restrict__ p, const float* __restrict__ x,
    const float* __restrict__ newp,
    const int* __restrict__ cidx, const int* __restrict__ nidx,
    const float* __restrict__ W1, const float* __restrict__ b1,
    float* __restrict__ part1) {
  extern __shared__ float smem[];
  v2f*   W1p = (v2f*)smem;             // [K1PAD/2][CMID] pairs (17408B)
  float* SS  = smem + K1PAD * CMID;    // [16][CMID]
  float* SQ  = SS + 16 * CMID;         // [16][CMID]
  stage_pairs(W1p, W1, K1PAD / 2, CMID, CIN + 3);
  __syncthreads();

  const int wid = threadIdx.x >> 5, lane = threadIdx.x & 31;
  const int half = lane >> 4, r = lane & 15;
  const int e0 = blockIdx.x * 128 + wid * 16;
  const int e  = e0 + r;                       // EEDGE % 128 == 0, no tail
  const int nb = nidx[e];
  const int cc = cidx[e];
  const float* xr = x + (size_t)nb * CIN;
  __builtin_prefetch(xr, 0, 1);
  const float dp0 = p[nb * 3 + 0] - newp[cc * 3 + 0];
  const float dp1 = p[nb * 3 + 1] - newp[cc * 3 + 1];
  const float dp2 = p[nb * 3 + 2] - newp[cc * 3 + 2];

  v8f acc[4];
  #pragma unroll
  for (int n = 0; n < 4; ++n) acc[n] = v8zero();

  #pragma unroll
  for (int kk = 0; kk < K1PAD / 4; ++kk) {
    const int kp = 2 * kk + half;
    const int ka = 2 * kp, kb = ka + 1;
    v2f a;
    a.x = (ka < 3) ? (ka == 0 ? dp0 : (ka == 1 ? dp1 : dp2))
                   : ((ka < 3 + CIN) ? xr[ka - 3] : 0.f);
    a.y = (kb < 3) ? (kb == 1 ? dp1 : dp2)
                   : ((kb < 3 + CIN) ? xr[kb - 3] : 0.f);
    #pragma unroll
    for (int n = 0; n < 4; ++n)
      acc[n] = wmma4(a, W1p[kp * CMID + n * 16 + r], acc[n]);
  }

  #pragma unroll
  for (int n = 0; n < 4; ++n) {
    const int col = n * 16 + r;
    const float bc = b1[col];
    float s = 0.f, q = 0.f;
    #pragma unroll
    for (int v = 0; v < 8; ++v) { float h = acc[n][v] + bc; s += h; q += h * h; }
    SS[(wid * 2 + half) * CMID + col] = s;
    SQ[(wid * 2 + half) * CMID + col] = q;
  }
  __syncthreads();
  if (threadIdx.x < CMID) {
    float s = 0.f, q = 0.f;
    for (int i = 0; i < 16; ++i) {
      s += SS[i * CMID + threadIdx.x];
      q += SQ[i * CMID + threadIdx.x];
    }
    part1[(size_t)blockIdx.x * (2 * CMID) + threadIdx.x] = s;
    part1[(size_t)blockIdx.x * (2 * CMID) + CMID + threadIdx.x] = q;
  }
}

// --------------------------- finalize BN stats -> per-channel affine
__global__ void k_stats(const float* __restrict__ part, int nblk, int C,
                        const float* __restrict__ g, const float* __restrict__ be,
                        float* __restrict__ scale, float* __restrict__ shift) {
  const int c = threadIdx.x;
  if (c >= C) return;
  double s = 0.0, q = 0.0;
  for (int i = 0; i < nblk; ++i) {
    s += (double)part[(size_t)i * (2 * C) + c];
    q += (double)part[(size_t)i * (2 * C) + C + c];
  }
  const double mu = s / (double)EEDGE;
  const double var = q / (double)EEDGE - mu * mu;
  const float rs = rsqrtf((float)var + BN_EPS);
  const float sc = rs * g[c];
  scale[c] = sc;
  shift[c] = be[c] - (float)mu * sc;
}

// ---- pass 2: recompute GEMM1, BN1+ReLU, GEMM2, BN2 stats, seg min/max
__global__ void __launch_bounds__(256) k_pass2(
    const float* __restrict__ p, const float* __restrict__ x,
    const float* __restrict__ newp,
    const int* __restrict__ cidx, const int* __restrict__ nidx,
    const float* __restrict__ W1, const float* __restrict__ b1,
    const float* __restrict__ scale1, const float* __restrict__ shift1,
    const float* __restrict__ W2, const float* __restrict__ b2,
    unsigned* __restrict__ pmax, unsigned* __restrict__ pmin,
    float* __restrict__ part2) {
  extern __shared__ float smem[];
  v2f*   W1p = (v2f*)smem;                       // [K1PAD/2][CMID] pairs
  v2f*   W2p = (v2f*)(smem + K1PAD * CMID);      // [CMID/2][COUT] pairs
  float* h1s = smem + K1PAD * CMID + CMID * COUT;// 8 waves * [16][CMID]
  float* SS  = h1s + 8 * 16 * CMID;              // [16][COUT]
  float* SQ  = SS + 16 * COUT;                   // [16][COUT]
  stage_pairs(W1p, W1, K1PAD / 2, CMID, CIN + 3);
  stage_pairs(W2p, W2, CMID / 2, COUT, CMID);
  __syncthreads();

  const int wid = threadIdx.x >> 5, lane = threadIdx.x & 31;
  const int half = lane >> 4, r = lane & 15;
  const int e0 = blockIdx.x * 128 + wid * 16;
  const int e  = e0 + r;
  const int nb = nidx[e];
  const int cc = cidx[e];
  const float* xr = x + (size_t)nb * CIN;
  __builtin_prefetch(xr, 0, 1);
  const float dp0 = p[nb * 3 + 0] - newp[cc * 3 + 0];
  const float dp1 = p[nb * 3 + 1] - newp[cc * 3 + 1];
  const float dp2 = p[nb * 3 + 2] - newp[cc * 3 + 2];

  // GEMM1 (same as pass 1)
  v8f acc1[4];
  #pragma unroll
  for (int n = 0; n < 4; ++n) acc1[n] = v8zero();
  #pragma unroll
  for (int kk = 0; kk < K1PAD / 4; ++kk) {
    const int kp = 2 * kk + half;
    const int ka = 2 * kp, kb = ka + 1;
    v2f a;
    a.x = (ka < 3) ? (ka == 0 ? dp0 : (ka == 1 ? dp1 : dp2))
                   : ((ka < 3 + CIN) ? xr[ka - 3] : 0.f);
    a.y = (kb < 3) ? (kb == 1 ? dp1 : dp2)
                   : ((kb < 3 + CIN) ? xr[kb - 3] : 0.f);
    #pragma unroll
    for (int n = 0; n < 4; ++n)
      acc1[n] = wmma4(a, W1p[kp * CMID + n * 16 + r], acc1[n]);
  }

  // BN1 + ReLU, redistribute C-layout -> A-layout through LDS
  float* h1w = h1s + wid * 16 * CMID;
  #pragma unroll
  for (int n = 0; n < 4; ++n) {
    const int col = n * 16 + r;
    const float bc = b1[col], sc = scale1[col], sh = shift1[col];
    #pragma unroll
    for (int v = 0; v < 8; ++v) {
      const float hp = acc1[n][v] + bc;
      h1w[(v + 8 * half) * CMID + col] = fmaxf(hp * sc + sh, 0.f);
    }
  }
  __syncthreads();

  // GEMM2: h1[16x64] @ W2[64x128]
  v8f acc2[8];
  #pragma unroll
  for (int n = 0; n < 8; ++n) acc2[n] = v8zero();
  #pragma unroll
  for (int kk = 0; kk < CMID / 4; ++kk) {
    const int kp = 2 * kk + half;
    const v2f a = *(const v2f*)(h1w + r * CMID + 2 * kp);  // aligned ds_load_b64
    #pragma unroll
    for (int n = 0; n < 8; ++n)
      acc2[n] = wmma4(a, W2p[kp * COUT + n * 16 + r], acc2[n]);
  }

  // per-row centroids for this lane's D rows
  int cE[8];
  #pragma unroll
  for (int v = 0; v < 8; ++v) cE[v] = cidx[e0 + v + 8 * half];

  // BN2 stats partials + deterministic segment max/min of pre-BN values
  #pragma unroll
  for (int n = 0; n < 8; ++n) {
    const int col = n * 16 + r;
    const float bc = b2[col];
    float s = 0.f, q = 0.f;
    #pragma unroll
    for (int v = 0; v < 8; ++v) {
      const float hv = acc2[n][v] + bc;
      s += hv; q += hv * hv;
      const unsigned o = ordf(hv);
      atomicMax(&pmax[(size_t)cE[v] * COUT + col], o);
      atomicMin(&pmin[(size_t)cE[v] * COUT + col], o);
    }
    SS[(wid * 2 + half) * COUT + col] = s;
    SQ[(wid * 2 + half) * COUT + col] = q;
  }
  __syncthreads();
  if (threadIdx.x < COUT) {
    float s = 0.f, q = 0.f;
    for (int i = 0; i < 16; ++i) {
      s += SS[i * COUT + threadIdx.x];
      q += SQ[i * COUT + threadIdx.x];
    }
    part2[(size_t)blockIdx.x * (2 * COUT) + threadIdx.x] = s;
    part2[(size_t)blockIdx.x * (2 * COUT) + COUT + threadIdx.x] = q;
  }
}

// ------------------------------------------------------- finalize
// pooled = bn2(segmax) (monotone affine: pick max or min by scale sign)
__global__ void __launch_bounds__(256) k_final(
    const unsigned* __restrict__ pmax, const unsigned* __restrict__ pmin,
    const float* __restrict__ scale2, const float* __restrict__ shift2,
    const float* __restrict__ identity, float* __restrict__ out) {
  const int i = blockIdx.x * blockDim.x + threadIdx.x;
  if (i >= MCENT * COUT) return;
  const int c = i & (COUT - 1);
  const unsigned um = pmax[i];
  float pooled;
  if (um == 0u) {                       // empty segment -> -inf -> 0
    pooled = 0.0f;
  } else {
    const float s = scale2[c];
    const float v = (s >= 0.f) ? unordf(um) : unordf(pmin[i]);
    pooled = v * s + shift2[c];
  }
  out[i] = fmaxf(pooled + identity[i], 0.0f);
}

// ---------------------------------------------------------------- host
extern "C" void kernel_launch(void* const* d_in, const int* in_sizes, int n_in,
                              void* d_out, int out_size, void* d_ws, size_t ws_size,
                              hipStream_t stream) {
  const float* p    = (const float*)d_in[0];
  const float* x    = (const float*)d_in[1];
  const int*   idx  = (const int*)d_in[2];
  const int*   cidx = (const int*)d_in[3];
  const int*   nidx = (const int*)d_in[4];
  const float* W1   = (const float*)d_in[5];
  const float* b1   = (const float*)d_in[6];
  const float* g1   = (const float*)d_in[7];
  const float* be1  = (const float*)d_in[8];
  const float* W2   = (const float*)d_in[9];
  const float* b2   = (const float*)d_in[10];
  const float* g2   = (const float*)d_in[11];
  const float* be2  = (const float*)d_in[12];
  const float* Wsk  = (const float*)d_in[13];
  const float* bsk  = (const float*)d_in[14];
  float* out = (float*)d_out;

  char* ws = (char*)d_ws;
  auto carve = [&](size_t bytes) -> char* {
    char* r = ws;
    ws += (bytes + 255) & ~(size_t)255;
    return r;
  };
  float*    newp   = (float*)carve((size_t)MCENT * 3 * 4);
  float*    iden   = (float*)carve((size_t)MCENT * COUT * 4);
  unsigned* pmax   = (unsigned*)carve((size_t)MCENT * COUT * 4);
  unsigned* pmin   = (unsigned*)carve((size_t)MCENT * COUT * 4);
  float*    part1  = (float*)carve((size_t)PASS_BLOCKS * 2 * CMID * 4);
  float*    part2  = (float*)carve((size_t)PASS_BLOCKS * 2 * COUT * 4);
  float*    scale1 = (float*)carve(CMID * 4);
  float*    shift1 = (float*)carve(CMID * 4);
  float*    scale2 = (float*)carve(COUT * 4);
  float*    shift2 = (float*)carve(COUT * 4);

  const int npool = MCENT * COUT;
  k_init<<<(npool + 255) / 256, 256, 0, stream>>>(pmax, pmin, npool);

  const int skip_blocks = (MCENT + 127) / 128;
  k_skip<<<skip_blocks, 256, CIN * COUT * 4, stream>>>(p, x, idx, Wsk, bsk, iden, newp);

  const size_t smem1 = (K1PAD * CMID + 2 * 16 * CMID) * 4;
  k_pass1<<<PASS_BLOCKS, 256, smem1, stream>>>(p, x, newp, cidx, nidx, W1, b1, part1);

  k_stats<<<1, CMID, 0, stream>>>(part1, PASS_BLOCKS, CMID, g1, be1, scale1, shift1);

  const size_t smem2 =
      (K1PAD * CMID + CMID * COUT + 8 * 16 * CMID + 2 * 16 * COUT) * 4;
  k_pass2<<<PASS_BLOCKS, 256, smem2, stream>>>(p, x, newp, cidx, nidx, W1, b1,
                                               scale1, shift1, W2, b2,
                                               pmax, pmin, part2);

  k_stats<<<1, COUT, 0, stream>>>(part2, PASS_BLOCKS, COUT, g2, be2, scale2, shift2);

  k_final<<<(npool + 255) / 256, 256, 0, stream>>>(pmax, pmin, scale2, shift2, iden, out);
}